// Capa_Harmonica_1_1443109012287
// MI455X (gfx1250) — compile-verified
//
#include <hip/hip_runtime.h>
#include <math.h>

typedef __attribute__((ext_vector_type(2))) float v2f;
typedef __attribute__((ext_vector_type(8))) float v8f;

#define N_LEN 4096
#define KER   256
#define C_OUT 64
#define C_IN  8
#define K_B   4
#define KDIM  4096          // 2 signals * C_IN * KER
#define MROWS 128           // 64 real-output rows + 64 imag-output rows
#define P_OUT 4353          // N + 2*pad - ker + 1
#define PSTR  4368          // padded row stride for zraw
#define NT    64            // output positions per block
#define ROWLEN (NT + KER)   // 320 floats per (signal, channel) slice in LDS

// ---------------------------------------------------------------------------
// Kernel 1: build the signed, stacked weight matrix W[128][4096].
//   row c<64  : [ wr[c] | -wi[c] ]  -> zraw_r = conv(zr,wr) - conv(zi,wi)
//   row c>=64 : [ wi[c] |  wr[c] ]  -> zraw_i = conv(zr,wi) + conv(zi,wr)
// ---------------------------------------------------------------------------
__global__ void build_weights(const float* __restrict__ A,
                              const float* __restrict__ beta,
                              const int* __restrict__ mptr,
                              float* __restrict__ W)
{
  int gid = blockIdx.x * blockDim.x + threadIdx.x;
  if (gid >= MROWS * KDIM) return;
  int c2  = gid / KDIM;
  int kap = gid - c2 * KDIM;
  int sh  = kap >> 11;          // 0: z_real half, 1: z_imag half
  int rem = kap & 2047;
  int ci  = rem >> 8;
  int t   = rem & 255;
  int c   = (c2 < C_OUT) ? c2 : (c2 - C_OUT);
  float absA  = fabsf(A[c * C_IN + ci]);
  float m     = (float)(*mptr);
  float omega = 6.28318530717958647692f * m / (float)N_LEN;
  float ang   = beta[c * C_IN + ci] + omega * (float)t;
  float wr = absA * cosf(ang);
  float wi = absA * sinf(ang);
  float val;
  if (c2 < C_OUT) val = (sh == 0) ? wr : -wi;
  else            val = (sh == 0) ? wi :  wr;
  W[gid] = val;
}

// ---------------------------------------------------------------------------
// Kernel 2: conv-as-GEMM via V_WMMA_F32_16X16X4_F32.
//   Per block (256 thr = 8 waves): one batch, NT=64 output positions,
//   all 128 output rows. Wave w -> M rows [16w,16w+16), four 16x16 C tiles
//   (one A float2 load feeds 4 WMMAs). B operands come from an LDS-staged,
//   zero-padded slice of z (im2col patches are sliding windows).
// ---------------------------------------------------------------------------
__global__ __launch_bounds__(256) void conv_wmma(const float* __restrict__ zr,
                                                 const float* __restrict__ zi,
                                                 const float* __restrict__ W,
                                                 float* __restrict__ zraw)
{
  __shared__ float zs[2 * C_IN * ROWLEN];      // 20 KB

  const int tid    = threadIdx.x;
  const int ptiles = (P_OUT + NT - 1) / NT;    // 69
  const int kb     = blockIdx.x / ptiles;
  const int pt     = blockIdx.x % ptiles;
  const int p0     = pt * NT;

  // Stage zero-padded input slice: zs[s][ci][o] = z_s[kb][ci][p0-256+o]
  for (int s = 0; s < 2; ++s) {
    const float* z = (s == 0) ? zr : zi;
    float* zdst = zs + s * (C_IN * ROWLEN);
    for (int idx = tid; idx < C_IN * ROWLEN; idx += 256) {
      int ci   = idx / ROWLEN;
      int o    = idx - ci * ROWLEN;
      int gsrc = p0 - KER + o;
      float v  = 0.0f;
      if (gsrc >= 0 && gsrc < N_LEN) v = z[(kb * C_IN + ci) * N_LEN + gsrc];
      zdst[idx] = v;
    }
  }
  __syncthreads();

  const int w    = tid >> 5;   // wave id 0..7  -> M rows 16w..16w+15
  const int lane = tid & 31;
  const int l    = lane & 15;  // A: row index, B/C: column index
  const int g    = lane >> 4;  // half-wave selector

  v8f acc0 = {};
  v8f acc1 = {};
  v8f acc2 = {};
  v8f acc3 = {};

  // A operand base: W[16w+l][... + 2g]  (lane layout: VGPRj = A[l][2g+j])
  const float* Wb = W + (size_t)(w * 16 + l) * KDIM + 2 * g;

  for (int s = 0; s < 2; ++s) {
    for (int ci = 0; ci < C_IN; ++ci) {
      const float* wrow = Wb + s * (C_IN * KER) + ci * KER;
      // per-lane sliding-window base: B[t'][16q+l] = zrow[16q + l + t0 + t' + ...]
      const float* zptr = zs + (s * C_IN + ci) * ROWLEN + l + 2 * g;
      __builtin_prefetch(wrow + KER, 0, 1);    // global_prefetch_b8: next row seg
      #pragma unroll 4
      for (int t0 = 0; t0 < KER; t0 += 4) {
        v2f a;
        a.x = wrow[t0];
        a.y = wrow[t0 + 1];
        v2f b0; b0.x = zptr[t0];      b0.y = zptr[t0 + 1];
        v2f b1; b1.x = zptr[t0 + 16]; b1.y = zptr[t0 + 17];
        v2f b2; b2.x = zptr[t0 + 32]; b2.y = zptr[t0 + 33];
        v2f b3; b3.x = zptr[t0 + 48]; b3.y = zptr[t0 + 49];
        acc0 = __builtin_amdgcn_wmma_f32_16x16x4_f32(
                   false, a, false, b0, (short)0, acc0, false, false);
        acc1 = __builtin_amdgcn_wmma_f32_16x16x4_f32(
                   false, a, false, b1, (short)0, acc1, false, false);
        acc2 = __builtin_amdgcn_wmma_f32_16x16x4_f32(
                   false, a, false, b2, (short)0, acc2, false, false);
        acc3 = __builtin_amdgcn_wmma_f32_16x16x4_f32(
                   false, a, false, b3, (short)0, acc3, false, false);
      }
    }
  }

  // C layout: VGPR r, lane g*16+n -> C[r+8g][n]
  float* zb = zraw + (size_t)kb * MROWS * PSTR;
  #pragma unroll
  for (int r = 0; r < 8; ++r) {
    int crow = w * 16 + r + 8 * g;
    float* row = zb + (size_t)crow * PSTR;
    int pa = p0 + l;
    int pb = p0 + 16 + l;
    int pc = p0 + 32 + l;
    int pd = p0 + 48 + l;
    if (pa < P_OUT) row[pa] = acc0[r];
    if (pb < P_OUT) row[pb] = acc1[r];
    if (pc < P_OUT) row[pc] = acc2[r];
    if (pd < P_OUT) row[pd] = acc3[r];
  }
}

// ---------------------------------------------------------------------------
// Kernel 3: alternating-sign comb over 32 shifts of 256, then sigmoid gate.
// ---------------------------------------------------------------------------
__global__ void comb_gate(const float* __restrict__ zraw,
                          const float* __restrict__ bias,
                          float* __restrict__ out)
{
  int gid = blockIdx.x * blockDim.x + threadIdx.x;
  if (gid >= K_B * C_OUT * N_LEN) return;
  int kb  = gid / (C_OUT * N_LEN);
  int rem = gid - kb * (C_OUT * N_LEN);
  int c   = rem / N_LEN;
  int mu  = rem - c * N_LEN;

  const float* zr_r = zraw + ((size_t)kb * MROWS + c) * PSTR;
  const float* zr_i = zraw + ((size_t)kb * MROWS + C_OUT + c) * PSTR;

  float sr = 0.0f, si = 0.0f;
  #pragma unroll
  for (int j = -15; j <= 16; ++j) {
    int ind = mu + KER * j;
    if (ind > 0 && ind < N_LEN + KER) {          // strict, matches reference
      float sg = (j & 1) ? 1.0f : -1.0f;         // (-1)^(1+j)
      sr += sg * zr_r[ind];
      si += sg * zr_i[ind];
    }
  }
  float mag  = sqrtf(sr * sr + si * si);
  float x    = mag + bias[c];
  float gate = (1.0f / (1.0f + expf(-x))) / (mag + 1e-5f);
  out[gid]                         = gate * sr;
  out[K_B * C_OUT * N_LEN + gid]   = gate * si;
}

// ---------------------------------------------------------------------------
extern "C" void kernel_launch(void* const* d_in, const int* in_sizes, int n_in,
                              void* d_out, int out_size, void* d_ws, size_t ws_size,
                              hipStream_t stream)
{
  const float* zr   = (const float*)d_in[0];
  const float* zi   = (const float*)d_in[1];
  const float* A    = (const float*)d_in[2];
  const float* beta = (const float*)d_in[3];
  const float* bias = (const float*)d_in[4];
  const int*   mp   = (const int*)d_in[5];

  // Workspace: W (2 MB) then zraw (4*128*4368 floats ~ 8.9 MB), ~11 MB total.
  float* W    = (float*)d_ws;
  float* zraw = W + (size_t)MROWS * KDIM;

  {
    int total = MROWS * KDIM;
    build_weights<<<(total + 255) / 256, 256, 0, stream>>>(A, beta, mp, W);
  }
  {
    int ptiles = (P_OUT + NT - 1) / NT;          // 69
    conv_wmma<<<K_B * ptiles, 256, 0, stream>>>(zr, zi, W, zraw);
  }
  {
    int total = K_B * C_OUT * N_LEN;
    comb_gate<<<(total + 255) / 256, 256, 0, stream>>>(zraw, bias, (float*)d_out);
  }
}